// Ansatz_44573170598544
// MI455X (gfx1250) — compile-verified
//
#include <hip/hip_runtime.h>
#include <math.h>

typedef float v2f __attribute__((ext_vector_type(2)));
typedef float v8f __attribute__((ext_vector_type(8)));

// ---------------- complex helpers (precompute kernel) ----------------
struct C2 { float r, i; };
__device__ __forceinline__ C2 cmul(C2 a, C2 b) {
    C2 o; o.r = a.r * b.r - a.i * b.i; o.i = a.r * b.i + a.i * b.r; return o;
}

__device__ void mm2(C2 o[2][2], const C2 A[2][2], const C2 Bm[2][2]) {
    for (int a = 0; a < 2; ++a)
        for (int c = 0; c < 2; ++c) {
            C2 acc = {0.f, 0.f};
            for (int k = 0; k < 2; ++k) {
                C2 t = cmul(A[a][k], Bm[k][c]);
                acc.r += t.r; acc.i += t.i;
            }
            o[a][c] = acc;
        }
}

__device__ void mm4(C2 o[4][4], const C2 A[4][4], const C2 Bm[4][4]) {
    for (int a = 0; a < 4; ++a)
        for (int c = 0; c < 4; ++c) {
            C2 acc = {0.f, 0.f};
            for (int k = 0; k < 4; ++k) {
                C2 t = cmul(A[a][k], Bm[k][c]);
                acc.r += t.r; acc.i += t.i;
            }
            o[a][c] = acc;
        }
}

// E = Rz(p2) @ Ry(p1) @ Rz(p0)
__device__ void euler2(C2 E[2][2], const float* p) {
    float c0 = cosf(0.5f * p[0]), s0 = sinf(0.5f * p[0]);
    float c1 = cosf(0.5f * p[1]), s1 = sinf(0.5f * p[1]);
    float c2 = cosf(0.5f * p[2]), s2 = sinf(0.5f * p[2]);
    C2 Rz0[2][2] = { { {c0, -s0}, {0.f, 0.f} }, { {0.f, 0.f}, {c0, s0} } };
    C2 Ry1[2][2] = { { {c1, 0.f}, {-s1, 0.f} }, { {s1, 0.f}, {c1, 0.f} } };
    C2 Rz2[2][2] = { { {c2, -s2}, {0.f, 0.f} }, { {0.f, 0.f}, {c2, s2} } };
    C2 T[2][2];
    mm2(T, Ry1, Rz0);
    mm2(E, Rz2, T);
}

// U_ent = rzz(t3) @ ryy(t2) @ rxx(t1) @ rzz(t0)
__device__ void entangler4(C2 U[4][4], const float* t) {
    const float dzz[4] = {1.f, -1.f, -1.f, 1.f};
    C2 rzz0[4][4] = {}, rzz3[4][4] = {}, rxx[4][4] = {}, ryy[4][4] = {};
    for (int k = 0; k < 4; ++k) {
        float a0 = 0.5f * t[0] * dzz[k];
        float a3 = 0.5f * t[3] * dzz[k];
        rzz0[k][k].r = cosf(a0); rzz0[k][k].i = -sinf(a0);   // exp(-i a0)
        rzz3[k][k].r = cosf(a3); rzz3[k][k].i = -sinf(a3);
    }
    float cx = cosf(0.5f * t[1]), sx = sinf(0.5f * t[1]);
    float cy = cosf(0.5f * t[2]), sy = sinf(0.5f * t[2]);
    for (int k = 0; k < 4; ++k) {
        rxx[k][k].r = cx; rxx[k][3 - k].i += -sx;            // cx*I - i*sx*XX
        ryy[k][k].r = cy;                                    // cy*I - i*sy*YY
    }
    ryy[0][3].i =  sy;  // YY[0][3] = -1 -> -i*sy*(-1) = +i*sy
    ryy[1][2].i = -sy;  // YY[1][2] = +1
    ryy[2][1].i = -sy;  // YY[2][1] = +1
    ryy[3][0].i =  sy;  // YY[3][0] = -1
    C2 T1[4][4], T2[4][4];
    mm4(T1, rxx, rzz0);
    mm4(T2, ryy, T1);
    mm4(U, rzz3, T2);
}

// One thread per (circuit, layer): build 4x4 layer unitary, store to ws.
// Layout: ws[((c*2+l)*16 + j*4 + a)*2 + {0=re,1=im}], 128 floats total.
__global__ void precompute_U_kernel(const float* __restrict__ psq,   // (2,2,4,3)
                                    const float* __restrict__ ptq,   // (2,2,4)
                                    float* __restrict__ Uws) {
    int t = threadIdx.x;
    if (t >= 4) return;
    int c = t >> 1, l = t & 1;
    C2 E[4][2][2];
    for (int q = 0; q < 4; ++q)
        euler2(E[q], psq + ((c * 2 + l) * 4 + q) * 3);
    C2 K1[4][4], K2[4][4];
    for (int a = 0; a < 2; ++a)
        for (int b = 0; b < 2; ++b)
            for (int cc = 0; cc < 2; ++cc)
                for (int d = 0; d < 2; ++d) {
                    K1[2 * a + b][2 * cc + d] = cmul(E[0][a][cc], E[1][b][d]);
                    K2[2 * a + b][2 * cc + d] = cmul(E[2][a][cc], E[3][b][d]);
                }
    C2 ent[4][4], T[4][4], U[4][4];
    entangler4(ent, ptq + (c * 2 + l) * 4);
    mm4(T, ent, K1);
    mm4(U, K2, T);
    float* dst = Uws + (c * 2 + l) * 32;
    for (int j = 0; j < 4; ++j)
        for (int a = 0; a < 4; ++a) {
            dst[(j * 4 + a) * 2 + 0] = U[j][a].r;
            dst[(j * 4 + a) * 2 + 1] = U[j][a].i;
        }
}

// ---------------- main fused kernel: one wave32 per batch item ----------------
#define WPB 8   // waves per block

__global__ __launch_bounds__(256) void ansatz_main(
    const float* __restrict__ x,     // (B, 32)
    const float* __restrict__ jp,    // (6,)
    const float* __restrict__ enc,   // (2,2,2)
    const float* __restrict__ encb,  // (2,2,2)
    const float* __restrict__ pc,    // (2,2)
    const float* __restrict__ kv,    // (8,2)
    const float* __restrict__ Uws,   // 128 floats
    float* __restrict__ out,         // (B,2) interleaved complex64
    int B) {
    __shared__ float smem[WPB][2][8][8][2];   // [wave][block][row n][col m][re/im]

    const int lane = threadIdx.x & 31;
    const int wave = threadIdx.x >> 5;
    int b = blockIdx.x * WPB + wave;
    if (b >= B) b = B - 1;                     // redundant recompute, identical writes

    const int circ = lane >> 4;                // 0: re-circuit, 1: im-circuit
    const int part = lane & 15;                // particle id (0..15)

    // ---- per-lane PQC for (particle=part, circuit=circ) ----
    const float r0 = x[b * 32 + part * 2 + 0];
    const float r1 = x[b * 32 + part * 2 + 1];
    const float TWO_PI = 6.28318530717958647692f;

    float pre[4] = {1.f, 0.f, 0.f, 0.f};
    float pim[4] = {0.f, 0.f, 0.f, 0.f};
    for (int l = 0; l < 2; ++l) {
        const int eo = (circ * 2 + l) * 2;
        float a0 = enc[eo + 0] * TWO_PI * r0 + encb[eo + 0];
        float a1 = enc[eo + 1] * TWO_PI * r1 + encb[eo + 1];
        float c0 = cosf(0.5f * a0), s0 = sinf(0.5f * a0);
        float c1 = cosf(0.5f * a1), s1 = sinf(0.5f * a1);
        // RX on qubit 0 (index pairs (0,2),(1,3)); RX amp: c real, s -> -i*sin
        for (int q = 0; q < 2; ++q) {
            int i0 = q, i1 = 2 + q;
            float xr = pre[i0], xi = pim[i0], yr = pre[i1], yi = pim[i1];
            pre[i0] = c0 * xr + s0 * yi;  pim[i0] = c0 * xi - s0 * yr;
            pre[i1] = c0 * yr + s0 * xi;  pim[i1] = c0 * yi - s0 * xr;
        }
        // RX on qubit 1 (index pairs (0,1),(2,3))
        for (int q = 0; q < 2; ++q) {
            int i0 = 2 * q, i1 = 2 * q + 1;
            float xr = pre[i0], xi = pim[i0], yr = pre[i1], yi = pim[i1];
            pre[i0] = c1 * xr + s1 * yi;  pim[i0] = c1 * xi - s1 * yr;
            pre[i1] = c1 * yr + s1 * xi;  pim[i1] = c1 * yi - s1 * xr;
        }
        // psi = U @ psi  (psi @ U.T in reference)
        const float* Um = Uws + (circ * 2 + l) * 32;
        float nr[4], ni[4];
        for (int j = 0; j < 4; ++j) {
            float ar = 0.f, ai = 0.f;
            for (int a = 0; a < 4; ++a) {
                float ur = Um[(j * 4 + a) * 2 + 0];
                float ui = Um[(j * 4 + a) * 2 + 1];
                ar += ur * pre[a] - ui * pim[a];
                ai += ur * pim[a] + ui * pre[a];
            }
            nr[j] = ar; ni[j] = ai;
        }
        for (int j = 0; j < 4; ++j) { pre[j] = nr[j]; pim[j] = ni[j]; }
    }
    float p0 = pre[0] * pre[0] + pim[0] * pim[0];
    float p1 = pre[1] * pre[1] + pim[1] * pim[1];
    float p2 = pre[2] * pre[2] + pim[2] * pim[2];
    float p3 = pre[3] * pre[3] + pim[3] * pim[3];
    float bf0 = p0 + p1 - p2 - p3;   // <Z0> -> shift dim 0
    float bf1 = p0 - p1 + p2 - p3;   // <Z1> -> shift dim 1

    // lanes<16 hold bf_re; fetch bf_im for the same particle from lane part+16
    float bfi0 = __shfl(bf0, part + 16, 32);
    float bfi1 = __shfl(bf1, part + 16, 32);
    float xcre0 = r0 + pc[0] * bf0;
    float xcre1 = r1 + pc[1] * bf1;
    float xcim0 = pc[2] * bfi0;
    float xcim1 = pc[3] * bfi1;

    // ---- phase = xc . k^T via V_WMMA_F32_16X16X4_F32 (K=2 zero-padded) ----
    // A 16x4 layout: vgpr0 = K0 (lanes0-15) / K2 (lanes16-31); vgpr1 = K1/K3.
    // B 4x16 layout: vgpr0 = row K0 (lanes0-15) / row K2; vgpr1 = K1/K3.
    const bool lo = (lane < 16);
    v2f Are, Aim, Bk;
    Are[0] = lo ? xcre0 : 0.f;  Are[1] = lo ? xcre1 : 0.f;
    Aim[0] = lo ? xcim0 : 0.f;  Aim[1] = lo ? xcim1 : 0.f;
    Bk[0] = (lo && part < 8) ? kv[part * 2 + 0] : 0.f;
    Bk[1] = (lo && part < 8) ? kv[part * 2 + 1] : 0.f;
    v8f zacc = {};
    v8f Dre = __builtin_amdgcn_wmma_f32_16x16x4_f32(false, Are, false, Bk,
                                                    (short)0, zacc, false, false);
    v8f Dim = __builtin_amdgcn_wmma_f32_16x16x4_f32(false, Aim, false, Bk,
                                                    (short)0, zacc, false, false);

    // D layout: vgpr v, lanes0-15 -> (particle v, kvec lane); lanes16-31 ->
    // (particle 8+v, kvec lane-16). Build Slater entries exp(i*(phr + i*phi)).
    if (part < 8) {
        for (int v = 0; v < 8; ++v) {
            float phr = Dre[v], phi = Dim[v];
            float e = expf(-phi);
            smem[wave][circ][v][part][0] = e * cosf(phr);
            smem[wave][circ][v][part][1] = e * sinf(phr);
        }
    }
    __syncthreads();

    // ---- 8x8 complex slogdet via LU w/ partial pivoting (lanes 0 and 16) ----
    float ldr = 0.f, ldi = 0.f;
    if (part == 0) {
        float (*Mm)[8][2] = smem[wave][circ];
        float logabs = 0.f, sre = 1.f, sim = 0.f;   // |det| log + unit phase
        for (int k = 0; k < 8; ++k) {
            int piv = k;
            float best = Mm[k][k][0] * Mm[k][k][0] + Mm[k][k][1] * Mm[k][k][1];
            for (int i2 = k + 1; i2 < 8; ++i2) {
                float mg = Mm[i2][k][0] * Mm[i2][k][0] + Mm[i2][k][1] * Mm[i2][k][1];
                if (mg > best) { best = mg; piv = i2; }
            }
            if (piv != k) {
                for (int j = k; j < 8; ++j) {
                    float t0 = Mm[k][j][0]; Mm[k][j][0] = Mm[piv][j][0]; Mm[piv][j][0] = t0;
                    float t1 = Mm[k][j][1]; Mm[k][j][1] = Mm[piv][j][1]; Mm[piv][j][1] = t1;
                }
                sre = -sre; sim = -sim;
            }
            float ur = Mm[k][k][0], ui = Mm[k][k][1];
            float m2 = ur * ur + ui * ui;
            logabs += 0.5f * logf(m2);
            float rinv = 1.0f / sqrtf(m2);
            float nsr = (sre * ur - sim * ui) * rinv;
            float nsi = (sre * ui + sim * ur) * rinv;
            sre = nsr; sim = nsi;
            float invm2 = 1.0f / m2;
            for (int i2 = k + 1; i2 < 8; ++i2) {
                float ar = Mm[i2][k][0], ai = Mm[i2][k][1];
                float fr = (ar * ur + ai * ui) * invm2;   // a / u
                float fi = (ai * ur - ar * ui) * invm2;
                for (int j = k + 1; j < 8; ++j) {
                    float br = Mm[k][j][0], bi = Mm[k][j][1];
                    Mm[i2][j][0] -= fr * br - fi * bi;
                    Mm[i2][j][1] -= fr * bi + fi * br;
                }
            }
        }
        ldr = logabs;
        ldi = atan2f(sim, sre);   // principal branch, matches log(sign)+logabs
    }

    // ---- Jastrow: 120 pairs split across the wave ----
    float jas = 0.f;
    for (int k = lane; k < 120; k += 32) {
        int i = 0, rem = k;
        while (rem >= 15 - i) { rem -= 15 - i; ++i; }
        int j = i + 1 + rem;
        float s = 0.f;
        for (int d = 0; d < 2; ++d) {
            float diff = x[b * 32 + i * 2 + d] - x[b * 32 + j * 2 + d];
            float md = (diff + 0.5f) - floorf(diff + 0.5f) - 0.5f;  // min image, L=1
            float ad = fabsf(md);
            float jv = ad * (1.0f - 2.0f * ad * ad * ad);           // JAS_C = 2
            s += jv * jv;
        }
        float t = sqrtf(s), tn = 1.f, acc = 0.f;
        for (int n = 0; n < 6; ++n) { tn *= t; acc += jp[n] * tn; }
        jas += acc;
    }
    for (int off = 16; off > 0; off >>= 1) jas += __shfl_xor(jas, off, 32);

    // ---- combine: lane 0 has up logdet, lane 16 has down ----
    float dnr = __shfl(ldr, 16, 32);
    float dni = __shfl(ldi, 16, 32);
    if (lane == 0) {
        out[2 * b + 0] = ldr + dnr + jas;
        out[2 * b + 1] = ldi + dni;
    }
}

extern "C" void kernel_launch(void* const* d_in, const int* in_sizes, int n_in,
                              void* d_out, int out_size, void* d_ws, size_t ws_size,
                              hipStream_t stream) {
    (void)n_in; (void)out_size; (void)ws_size;
    const float* x    = (const float*)d_in[0];
    const float* jp   = (const float*)d_in[1];
    const float* psq  = (const float*)d_in[2];
    const float* ptq  = (const float*)d_in[3];
    const float* enc  = (const float*)d_in[4];
    const float* encb = (const float*)d_in[5];
    const float* pc   = (const float*)d_in[6];
    const float* kv   = (const float*)d_in[7];
    float* out = (float*)d_out;
    float* Uws = (float*)d_ws;

    const int B = in_sizes[0] / 32;   // N*SDIM = 32 floats per sample

    precompute_U_kernel<<<1, 4, 0, stream>>>(psq, ptq, Uws);
    const int nblk = (B + WPB - 1) / WPB;
    ansatz_main<<<nblk, 32 * WPB, 0, stream>>>(x, jp, enc, encb, pc, kv, Uws, out, B);
}